// ResetDNC_63376537420220
// MI455X (gfx1250) — compile-verified
//
#include <hip/hip_runtime.h>
#include <math.h>

// ---------------- sizes ----------------
#define BSZ 4
#define XD  1024
#define HD  256
#define LD  4
#define VTD 4096
#define WDM 64
#define RD  8
#define ND  4096
#define ETD 747
#define IND 2048
#define EPSF 1e-8f

typedef __attribute__((ext_vector_type(2))) float v2f;
typedef __attribute__((ext_vector_type(8))) float v8f;

__device__ __forceinline__ float sigm_(float x){ return 1.f/(1.f+expf(-x)); }
__device__ __forceinline__ float logsig_(float x){ return fminf(x,0.f) - log1pf(expf(-fabsf(x))); }

// =====================================================================
// WMMA f32 GEMM:  Out[4,NC] = A[4,KK] @ B[KK,NC] (+ addend[4,NC])
// A padded to M=16 (zero via mask multiply, no divergent loads).
// One wave per 16-column tile. V_WMMA_F32_16X16X4_F32.
// Templated on K and N so all strides are compile-time constants.
// =====================================================================
template <int KK, int NC>
__global__ __launch_bounds__(256) void dnc_wmma_gemm(const float* __restrict__ A,
                                                     const float* __restrict__ B,
                                                     const float* __restrict__ addend,
                                                     float* __restrict__ Out) {
    const int lane  = threadIdx.x & 31;
    const int wave  = blockIdx.x * (blockDim.x >> 5) + (threadIdx.x >> 5);
    const int col0  = wave * 16;
    const int m     = lane & 15;          // A row (M); also N within tile for B/C
    const int khalf = (lane >> 4) * 2;    // 0 or 2
    const int col   = col0 + m;
    const bool colok = (col < NC);
    const int colc  = colok ? col : (NC - 1);     // clamped load address
    const float amask = (m < BSZ) ? 1.f : 0.f;    // zero-pad rows 4..15

    const float* Arow = A + (m & 3) * KK + khalf; // clamped row, always in-bounds
    const float* Bx   = B + (size_t)khalf * NC + colc;
    const float* By   = Bx + NC;

    v8f acc = {};
    #pragma unroll 8
    for (int k = 0; k < KK; k += 4) {
        v2f a = *(const v2f*)(Arow + k);          // single b64 load
        a.x *= amask; a.y *= amask;
        v2f b;
        b.x = *Bx;  b.y = *By;
        Bx += 4 * NC;  By += 4 * NC;
        // 8 args: (neg_a, A, neg_b, B, c_mod, C, reuse_a, reuse_b)
        acc = __builtin_amdgcn_wmma_f32_16x16x4_f32(false, a, false, b,
                                                    (short)0, acc, false, false);
    }
    // C/D layout: acc[v] holds row M = v + 8*(lane>>4), col = col0 + (lane&15)
    const int rowbase = (lane >> 4) * 8;
    if (colok && rowbase == 0) {                  // only rows 0..7 half can contain rows <4
        #pragma unroll
        for (int v = 0; v < BSZ; ++v) {
            float val = acc[v];
            if (addend) val += addend[(size_t)v * NC + col];
            Out[(size_t)v * NC + col] = val;
        }
    }
}

// =====================================================================
// LSTM controller helpers
// =====================================================================
__global__ void dnc_build_s(const float* __restrict__ x_in, const float* __restrict__ lrv,
                            const float* __restrict__ ctrl_h, const float* __restrict__ prev,
                            int layer, float* __restrict__ s) {
    int id = blockIdx.x * blockDim.x + threadIdx.x;       // 4*2048
    if (id >= BSZ * IND) return;
    int b = id >> 11, k = id & 2047;
    float v;
    if (k < XD)              v = x_in[b * XD + k];
    else if (k < XD + 512)   v = lrv[b * 512 + (k - XD)];                 // last_read_vec flat [w][r]
    else if (k < XD + 768)   v = ctrl_h[(b * LD + layer) * HD + (k - XD - 512)];
    else                     v = (layer == 0) ? 0.f : prev[b * HD + (k - XD - 768)];
    s[id] = v;
}

__global__ void dnc_gate_combine(const float* gi, const float* gf, const float* go, const float* gs,
                                 const float* b_i, const float* b_f, const float* b_o, const float* b_s,
                                 const float* ctrl_c, int layer,
                                 float* __restrict__ prev, float* __restrict__ flat) {
    int id = blockIdx.x * blockDim.x + threadIdx.x;       // 4*256
    if (id >= BSZ * HD) return;
    int b = id >> 8, j = id & 255;
    float ig = sigm_(gi[id] + b_i[layer * HD + j]);
    float fg = sigm_(gf[id] + b_f[layer * HD + j]);
    float og = sigm_(go[id] + b_o[layer * HD + j]);
    float c  = fg * ctrl_c[(b * LD + layer) * HD + j] + ig * tanhf(gs[id] + b_s[layer * HD + j]);
    float h  = og * tanhf(c);
    prev[id] = h;
    flat[b * (LD * HD) + layer * HD + j] = h;
}

// =====================================================================
// Usage: u[b,n] = (lu + lww - lu*lww) * prod_r(1 - free_r*lrw[b,n,r]) * nf
// =====================================================================
__global__ void dnc_usage(const float* __restrict__ iv, const float* __restrict__ lrw,
                          const float* __restrict__ lu, const float* __restrict__ lww,
                          const float* __restrict__ nf, float* __restrict__ u) {
    int id = blockIdx.x * blockDim.x + threadIdx.x;       // 4*4096
    if (id >= BSZ * ND) return;
    int b = id >> 12, n = id & (ND - 1);
    const float* ivb = iv + b * ETD;
    float psi = 1.f;
    #pragma unroll
    for (int r = 0; r < RD; ++r)
        psi *= (1.f - sigm_(ivb[713 + r]) * lrw[((size_t)b * ND + n) * RD + r]);
    float a = lu[id], w = lww[id];
    u[id] = (a + w - a * w) * psi * nf[0];
}

// =====================================================================
// Allocation: bitonic sort (ascending) of u, product-scan, scatter.
// One block of 1024 threads per batch.
// =====================================================================
__global__ __launch_bounds__(1024) void dnc_alloc_sort(const float* __restrict__ u,
                                                       float* __restrict__ alloc_w) {
    __shared__ float key[ND];
    __shared__ int   idx[ND];
    __shared__ float cp[ND];
    int b = blockIdx.x, t = threadIdx.x;
    for (int k = t; k < ND; k += 1024) { key[k] = u[b * ND + k]; idx[k] = k; }
    __syncthreads();
    for (int size = 2; size <= ND; size <<= 1) {
        for (int stride = size >> 1; stride > 0; stride >>= 1) {
            for (int n = t; n < ND / 2; n += 1024) {
                int pos = 2 * n - (n & (stride - 1));
                int a = pos, c = pos + stride;
                bool desc = (a & size) != 0;
                float ka = key[a], kc = key[c];
                bool sw = desc ? (ka < kc) : (ka > kc);
                if (sw) {
                    key[a] = kc; key[c] = ka;
                    int ti = idx[a]; idx[a] = idx[c]; idx[c] = ti;
                }
            }
            __syncthreads();
        }
    }
    // inclusive product scan of sorted keys
    for (int k = t; k < ND; k += 1024) cp[k] = key[k];
    __syncthreads();
    for (int off = 1; off < ND; off <<= 1) {
        float tmp[ND / 1024];
        for (int k = t, q = 0; k < ND; k += 1024, ++q)
            tmp[q] = (k >= off) ? cp[k - off] * cp[k] : cp[k];
        __syncthreads();
        for (int k = t, q = 0; k < ND; k += 1024, ++q) cp[k] = tmp[q];
        __syncthreads();
    }
    for (int k = t; k < ND; k += 1024) {
        float cpe = (k == 0) ? 1.f : cp[k - 1];
        alloc_w[b * ND + idx[k]] = (1.f - key[k]) * cpe;
    }
}

// =====================================================================
// Write weighting: cosine content softmax over N, blend with alloc_w.
// =====================================================================
__global__ __launch_bounds__(1024) void dnc_write_weight(const float* __restrict__ memory,
                                                         const float* __restrict__ iv,
                                                         const float* __restrict__ alloc_w,
                                                         float* __restrict__ ww) {
    __shared__ float sc[ND];
    __shared__ float red[1024];
    int b = blockIdx.x, t = threadIdx.x;
    const float* ivb = iv + b * ETD;
    float wkn = 0.f;
    for (int w = 0; w < WDM; ++w) { float v = ivb[520 + w]; wkn += v * v; }
    wkn = sqrtf(wkn) + EPSF;
    float wb = 1.f - logsig_(ivb[584]);
    float lmax = -3.4e38f;
    for (int n = t; n < ND; n += 1024) {
        const float* row = memory + ((size_t)b * ND + n) * WDM;
        float nm = 0.f, dot = 0.f;
        for (int w = 0; w < WDM; ++w) { float m = row[w]; nm += m * m; dot += m * ivb[520 + w]; }
        float s = wb * dot / ((sqrtf(nm) + EPSF) * wkn);
        sc[n] = s; lmax = fmaxf(lmax, s);
    }
    red[t] = lmax; __syncthreads();
    for (int s = 512; s > 0; s >>= 1) { if (t < s) red[t] = fmaxf(red[t], red[t + s]); __syncthreads(); }
    float smax = red[0]; __syncthreads();
    float lsum = 0.f;
    for (int n = t; n < ND; n += 1024) { float e = expf(sc[n] - smax); sc[n] = e; lsum += e; }
    red[t] = lsum; __syncthreads();
    for (int s = 512; s > 0; s >>= 1) { if (t < s) red[t] += red[t + s]; __syncthreads(); }
    float ssum = red[0]; __syncthreads();
    float ag = sigm_(ivb[721]), wg = sigm_(ivb[722]);
    for (int n = t; n < ND; n += 1024)
        ww[b * ND + n] = wg * (ag * alloc_w[b * ND + n] + (1.f - ag) * sc[n] / ssum);
}

// =====================================================================
// Memory update: new_mem = mem*(1 - ww*erase) + ww*write_vec
// =====================================================================
__global__ void dnc_mem_update(const float* __restrict__ memory, const float* __restrict__ iv,
                               const float* __restrict__ ww, float* __restrict__ new_mem) {
    size_t id = (size_t)blockIdx.x * blockDim.x + threadIdx.x;   // 4*4096*64
    if (id >= (size_t)BSZ * ND * WDM) return;
    int b = (int)(id >> 18), n = (int)((id >> 6) & (ND - 1)), w = (int)(id & 63);
    const float* ivb = iv + b * ETD;
    float e  = sigm_(ivb[585 + w]);
    float wv = ivb[649 + w];
    float wn = ww[b * ND + n];
    new_mem[id] = memory[id] * (1.f - wn * e) + wn * wv;
}

// =====================================================================
// fw[b,i,r] = sum_j L[b,i,j]*lrw[b,j,r],  L computed on the fly from tml.
// LDS-tiled 64x64; block = (i-tile, b); 256 threads own (i_local, r/r+4).
// Prefetch next j-tile while computing current one.
// =====================================================================
__global__ void dnc_fw(const float* __restrict__ tml, const float* __restrict__ ww,
                       const float* __restrict__ prec, const float* __restrict__ lrw,
                       const float* __restrict__ nf, float* __restrict__ fw) {
    __shared__ float tile[64][65];
    __shared__ float lrw_s[64][8];
    __shared__ float wwj_s[64];
    __shared__ float pj_s[64];
    int b = blockIdx.y, i0 = blockIdx.x * 64, t = threadIdx.x;
    int il = t & 63, r0 = t >> 6;
    int ii = i0 + il;
    float wwi = ww[b * ND + ii];
    float nfv = nf[0];
    float acc0 = 0.f, acc1 = 0.f;
    for (int j0 = 0; j0 < ND; j0 += 64) {
        for (int e = t; e < 64 * 64; e += 256) {
            int ti = e >> 6, tj = e & 63;
            tile[ti][tj] = tml[((size_t)b * ND + (i0 + ti)) * ND + (j0 + tj)];
        }
        if (j0 + 64 < ND) {   // prefetch next 64x64 tile (global_prefetch_b8)
            int ti = t >> 2, seg = (t & 3) << 4;
            __builtin_prefetch(&tml[((size_t)b * ND + (i0 + ti)) * ND + (j0 + 64) + seg], 0, 0);
        }
        for (int e = t; e < 64 * 8; e += 256)
            lrw_s[e >> 3][e & 7] = lrw[((size_t)b * ND + j0 + (e >> 3)) * RD + (e & 7)];
        if (t < 64) { wwj_s[t] = ww[b * ND + j0 + t]; pj_s[t] = prec[b * ND + j0 + t]; }
        __syncthreads();
        for (int tj = 0; tj < 64; ++tj) {
            int j = j0 + tj;
            float Lij = (1.f - wwi - wwj_s[tj]) * tile[il][tj] + wwi * pj_s[tj];
            Lij = (j == ii) ? 0.f : Lij * nfv;
            acc0 += Lij * lrw_s[tj][r0];
            acc1 += Lij * lrw_s[tj][r0 + 4];
        }
        __syncthreads();
    }
    fw[((size_t)b * ND + ii) * RD + r0]     = acc0;
    fw[((size_t)b * ND + ii) * RD + r0 + 4] = acc1;
}

// bw[b,i,r] = sum_j L[b,j,i]*lrw[b,j,r]   (transposed access, tile loaded coalesced)
__global__ void dnc_bw(const float* __restrict__ tml, const float* __restrict__ ww,
                       const float* __restrict__ prec, const float* __restrict__ lrw,
                       const float* __restrict__ nf, float* __restrict__ bw) {
    __shared__ float tile[64][65];
    __shared__ float lrw_s[64][8];
    __shared__ float wwj_s[64];
    int b = blockIdx.y, i0 = blockIdx.x * 64, t = threadIdx.x;
    int il = t & 63, r0 = t >> 6;
    int ii = i0 + il;
    float wwi = ww[b * ND + ii];
    float pi  = prec[b * ND + ii];
    float nfv = nf[0];
    float acc0 = 0.f, acc1 = 0.f;
    for (int j0 = 0; j0 < ND; j0 += 64) {
        for (int e = t; e < 64 * 64; e += 256) {
            int tj = e >> 6, ti = e & 63;
            tile[tj][ti] = tml[((size_t)b * ND + (j0 + tj)) * ND + (i0 + ti)];
        }
        if (j0 + 64 < ND) {   // prefetch next tile
            int tj = t >> 2, seg = (t & 3) << 4;
            __builtin_prefetch(&tml[((size_t)b * ND + (j0 + 64 + tj)) * ND + i0 + seg], 0, 0);
        }
        for (int e = t; e < 64 * 8; e += 256)
            lrw_s[e >> 3][e & 7] = lrw[((size_t)b * ND + j0 + (e >> 3)) * RD + (e & 7)];
        if (t < 64) wwj_s[t] = ww[b * ND + j0 + t];
        __syncthreads();
        for (int tj = 0; tj < 64; ++tj) {
            int j = j0 + tj;
            float wwj = wwj_s[tj];
            float Lji = (1.f - wwj - wwi) * tile[tj][il] + wwj * pi;
            Lji = (j == ii) ? 0.f : Lji * nfv;
            acc0 += Lji * lrw_s[tj][r0];
            acc1 += Lji * lrw_s[tj][r0 + 4];
        }
        __syncthreads();
    }
    bw[((size_t)b * ND + ii) * RD + r0]     = acc0;
    bw[((size_t)b * ND + ii) * RD + r0 + 4] = acc1;
}

// =====================================================================
// Read weighting: cosine softmax over N per read head + mode mixing.
// =====================================================================
__global__ __launch_bounds__(1024) void dnc_read_weight(const float* __restrict__ new_mem,
                                                        const float* __restrict__ iv,
                                                        const float* __restrict__ fw,
                                                        const float* __restrict__ bw,
                                                        float* __restrict__ rc,
                                                        float* __restrict__ rw) {
    __shared__ float rknorm[RD];
    __shared__ float red[1024];
    int b = blockIdx.x, t = threadIdx.x;
    const float* ivb = iv + b * ETD;
    if (t < RD) {
        float s = 0.f;
        for (int w = 0; w < WDM; ++w) { float v = ivb[w * RD + t]; s += v * v; }
        rknorm[t] = sqrtf(s) + EPSF;
    }
    __syncthreads();
    float lmax[RD];
    #pragma unroll
    for (int r = 0; r < RD; ++r) lmax[r] = -3.4e38f;
    for (int n = t; n < ND; n += 1024) {
        const float* row = new_mem + ((size_t)b * ND + n) * WDM;
        float nm = 0.f, dot[RD];
        #pragma unroll
        for (int r = 0; r < RD; ++r) dot[r] = 0.f;
        for (int w = 0; w < WDM; ++w) {
            float m = row[w]; nm += m * m;
            #pragma unroll
            for (int r = 0; r < RD; ++r) dot[r] += m * ivb[w * RD + r];
        }
        nm = sqrtf(nm) + EPSF;
        #pragma unroll
        for (int r = 0; r < RD; ++r) {
            float beta = 1.f - logsig_(ivb[512 + r]);
            float sc = beta * dot[r] / (nm * rknorm[r]);
            rc[((size_t)b * ND + n) * RD + r] = sc;
            lmax[r] = fmaxf(lmax[r], sc);
        }
    }
    float smax[RD], ssum[RD];
    for (int r = 0; r < RD; ++r) {
        red[t] = lmax[r]; __syncthreads();
        for (int s = 512; s > 0; s >>= 1) { if (t < s) red[t] = fmaxf(red[t], red[t + s]); __syncthreads(); }
        smax[r] = red[0]; __syncthreads();
    }
    float lsum[RD];
    #pragma unroll
    for (int r = 0; r < RD; ++r) lsum[r] = 0.f;
    for (int n = t; n < ND; n += 1024) {
        #pragma unroll
        for (int r = 0; r < RD; ++r) {
            float e = expf(rc[((size_t)b * ND + n) * RD + r] - smax[r]);
            rc[((size_t)b * ND + n) * RD + r] = e;
            lsum[r] += e;
        }
    }
    for (int r = 0; r < RD; ++r) {
        red[t] = lsum[r]; __syncthreads();
        for (int s = 512; s > 0; s >>= 1) { if (t < s) red[t] += red[t + s]; __syncthreads(); }
        ssum[r] = red[0]; __syncthreads();
    }
    float m0[RD], m1[RD], m2[RD];
    #pragma unroll
    for (int r = 0; r < RD; ++r) {
        float a = ivb[723 + r * 3], bb = ivb[723 + r * 3 + 1], c = ivb[723 + r * 3 + 2];
        float mx = fmaxf(a, fmaxf(bb, c));
        float ea = expf(a - mx), eb = expf(bb - mx), ec = expf(c - mx);
        float es = ea + eb + ec;
        m0[r] = ea / es; m1[r] = eb / es; m2[r] = ec / es;
    }
    for (int n = t; n < ND; n += 1024) {
        #pragma unroll
        for (int r = 0; r < RD; ++r) {
            size_t o = ((size_t)b * ND + n) * RD + r;
            float rcv = rc[o] / ssum[r];
            rw[o] = m0[r] * bw[o] + m1[r] * rcv + m2[r] * fw[o];
        }
    }
}

// =====================================================================
// read_vecs[b,w,r] = sum_n new_mem[b,n,w]*rw[b,n,r]; flat layout [w][r]
// =====================================================================
__global__ void dnc_read_vecs(const float* __restrict__ new_mem, const float* __restrict__ rw,
                              float* __restrict__ rv) {
    __shared__ float red[4][WDM][RD];
    int b = blockIdx.x, t = threadIdx.x;   // 256
    int w = t & 63, stripe = t >> 6;       // 4 stripes
    float acc[RD];
    #pragma unroll
    for (int r = 0; r < RD; ++r) acc[r] = 0.f;
    for (int n = stripe; n < ND; n += 4) {
        float m = new_mem[((size_t)b * ND + n) * WDM + w];
        #pragma unroll
        for (int r = 0; r < RD; ++r) acc[r] += m * rw[((size_t)b * ND + n) * RD + r];
    }
    #pragma unroll
    for (int r = 0; r < RD; ++r) red[stripe][w][r] = acc[r];
    __syncthreads();
    if (stripe == 0) {
        #pragma unroll
        for (int r = 0; r < RD; ++r)
            rv[b * (WDM * RD) + w * RD + r] =
                red[0][w][r] + red[1][w][r] + red[2][w][r] + red[3][w][r];
    }
}

// =====================================================================
extern "C" void kernel_launch(void* const* d_in, const int* in_sizes, int n_in,
                              void* d_out, int out_size, void* d_ws, size_t ws_size,
                              hipStream_t stream) {
    const float* x_in   = (const float*)d_in[0];
    const float* prec   = (const float*)d_in[1];
    const float* tml    = (const float*)d_in[2];
    const float* memory = (const float*)d_in[3];
    const float* lrw    = (const float*)d_in[4];
    const float* lu     = (const float*)d_in[5];
    const float* lww    = (const float*)d_in[6];
    const float* nf     = (const float*)d_in[7];
    const float* lrv    = (const float*)d_in[8];
    const float* ctrl_h = (const float*)d_in[9];
    const float* ctrl_c = (const float*)d_in[10];
    const float* Wi     = (const float*)d_in[11];
    const float* Wf     = (const float*)d_in[12];
    const float* Wo     = (const float*)d_in[13];
    const float* Ws     = (const float*)d_in[14];
    const float* b_i    = (const float*)d_in[15];
    const float* b_f    = (const float*)d_in[16];
    const float* b_o    = (const float*)d_in[17];
    const float* b_s    = (const float*)d_in[18];
    const float* W_y    = (const float*)d_in[19];
    const float* W_E    = (const float*)d_in[20];
    const float* W_r    = (const float*)d_in[21];
    float* out = (float*)d_out;

    float* ws   = (float*)d_ws;
    float* sbuf = ws;               // 8192
    float* gi   = sbuf + 8192;      // 1024
    float* gf   = gi + 1024;
    float* go   = gf + 1024;
    float* gs   = go + 1024;
    float* prev = gs + 1024;        // 1024
    float* flat = prev + 1024;      // 4096
    float* ybuf = flat + 4096;      // 16384
    float* iv   = ybuf + 16384;     // 3072 (747*4 used)
    float* u    = iv + 3072;        // 16384
    float* aw   = u + 16384;        // 16384
    float* wwb  = aw + 16384;       // 16384
    float* nmb  = wwb + 16384;      // 1048576
    float* fwb  = nmb + 1048576;    // 131072
    float* bwb  = fwb + 131072;     // 131072
    float* rcb  = bwb + 131072;     // 131072
    float* rwb  = rcb + 131072;     // 131072
    float* rvb  = rwb + 131072;     // 2048

    // ---- controller LSTM (4 sequential layers, WMMA gate GEMMs) ----
    for (int l = 0; l < LD; ++l) {
        dnc_build_s<<<32, 256, 0, stream>>>(x_in, lrv, ctrl_h, prev, l, sbuf);
        const size_t wo = (size_t)l * IND * HD;
        dnc_wmma_gemm<IND, HD><<<2, 256, 0, stream>>>(sbuf, Wi + wo, nullptr, gi);
        dnc_wmma_gemm<IND, HD><<<2, 256, 0, stream>>>(sbuf, Wf + wo, nullptr, gf);
        dnc_wmma_gemm<IND, HD><<<2, 256, 0, stream>>>(sbuf, Wo + wo, nullptr, go);
        dnc_wmma_gemm<IND, HD><<<2, 256, 0, stream>>>(sbuf, Ws + wo, nullptr, gs);
        dnc_gate_combine<<<4, 256, 0, stream>>>(gi, gf, go, gs, b_i, b_f, b_o, b_s,
                                                ctrl_c, l, prev, flat);
    }

    // ---- output + interface projections (WMMA) ----
    dnc_wmma_gemm<LD * HD, VTD><<<32, 256, 0, stream>>>(flat, W_y, nullptr, ybuf);
    dnc_wmma_gemm<LD * HD, ETD><<<6, 256, 0, stream>>>(flat, W_E, nullptr, iv);

    // ---- usage, allocation (sort), write weighting, memory update ----
    dnc_usage<<<64, 256, 0, stream>>>(iv, lrw, lu, lww, nf, u);
    dnc_alloc_sort<<<BSZ, 1024, 0, stream>>>(u, aw);
    dnc_write_weight<<<BSZ, 1024, 0, stream>>>(memory, iv, aw, wwb);
    dnc_mem_update<<<4096, 256, 0, stream>>>(memory, iv, wwb, nmb);

    // ---- temporal links fused with read-weight matvecs (stream tml) ----
    dnc_fw<<<dim3(ND / 64, BSZ), 256, 0, stream>>>(tml, wwb, prec, lrw, nf, fwb);
    dnc_bw<<<dim3(ND / 64, BSZ), 256, 0, stream>>>(tml, wwb, prec, lrw, nf, bwb);

    // ---- read weighting + read vectors ----
    dnc_read_weight<<<BSZ, 1024, 0, stream>>>(nmb, iv, fwb, bwb, rcb, rwb);
    dnc_read_vecs<<<BSZ, 256, 0, stream>>>(nmb, rwb, rvb);

    // ---- out = y + read_vecs_flat @ W_r (WMMA, fused add) ----
    dnc_wmma_gemm<WDM * RD, VTD><<<32, 256, 0, stream>>>(rvb, W_r, ybuf, out);
}